// DistanceLoss_43739946942958
// MI455X (gfx1250) — compile-verified
//
#include <hip/hip_runtime.h>
#include <hip/hip_bf16.h>

typedef __attribute__((ext_vector_type(2))) float v2f;
typedef __attribute__((ext_vector_type(8))) float v8f;

#define NN 512
#define DD 512
#define KC 32              // k-chunk staged in LDS
#define LROW 36            // padded LDS row stride in floats (144B, 16B-aligned)

// workspace layout (float offsets)
#define WS_TOT 0            // [2] per-matrix total sums
#define WS_SQ  16           // [2][512] row squared norms
#define WS_P   2048         // [2][512*512] pairwise smooth-l1 sums

// ---------------------------------------------------------------- init
__global__ void init_kernel(float* __restrict__ ws, float* __restrict__ out) {
    ws[WS_TOT + 0] = 0.f;
    ws[WS_TOT + 1] = 0.f;
    out[0] = 0.f;
}

// ---------------------------------------------------------------- row norms
__global__ __launch_bounds__(128) void rownorm_kernel(const float* __restrict__ T,
                                                      const float* __restrict__ S,
                                                      float* __restrict__ ws) {
    const int b   = blockIdx.x;          // 0..1023
    const int mat = b >> 9;
    const int row = b & 511;
    const float* X = (mat ? S : T) + (size_t)row * DD;
    float s = 0.f;
    for (int k = threadIdx.x; k < DD; k += 128) s = fmaf(X[k], X[k], s);
    __shared__ float red[128];
    red[threadIdx.x] = s;
    __syncthreads();
    for (int off = 64; off > 0; off >>= 1) {
        if (threadIdx.x < off) red[threadIdx.x] += red[threadIdx.x + off];
        __syncthreads();
    }
    if (threadIdx.x == 0) ws[WS_SQ + mat * NN + row] = red[0];
}

// ---------------------------------------------------------------- TDM staging
#if __has_builtin(__builtin_amdgcn_tensor_load_to_lds)
#define HAVE_TDM 1
typedef unsigned int u32x4 __attribute__((ext_vector_type(4)));
typedef int          i32x4 __attribute__((ext_vector_type(4)));
typedef int          i32x8 __attribute__((ext_vector_type(8)));

// DMA one 16-row x 32-float tile (row stride 512 floats) from global into LDS,
// inserting 4 pad DWORDs after every 32 data DWORDs (-> LROW=36 layout).
__device__ __forceinline__ void tdm_load_tile(const float* gsrc, unsigned lds_byte_addr) {
    const unsigned long long ga = (unsigned long long)(size_t)gsrc;
    u32x4 g0;
    g0[0] = (unsigned)__builtin_amdgcn_readfirstlane(1);                 // count=1, user desc
    g0[1] = (unsigned)__builtin_amdgcn_readfirstlane((int)lds_byte_addr);// lds_addr
    g0[2] = (unsigned)__builtin_amdgcn_readfirstlane((int)(unsigned)ga); // global_addr[31:0]
    g0[3] = (unsigned)__builtin_amdgcn_readfirstlane(
                (int)(((unsigned)(ga >> 32) & 0x01FFFFFFu) | (2u << 30)));// ga[56:32] | type=2
    i32x8 g1;
    // data_size=4B(2)<<16 | pad_enable<<20 | pad_interval(32dw=4)<<22 | pad_amount(4dw=3)<<25
    g1[0] = __builtin_amdgcn_readfirstlane((int)((2u << 16) | (1u << 20) |
                                                 (4u << 22) | (3u << 25)));
    g1[1] = __builtin_amdgcn_readfirstlane((int)(512u << 16)); // tensor_dim0[15:0] in [31:16]
    g1[2] = __builtin_amdgcn_readfirstlane((int)(512u << 16)); // dim0 hi=0 | tensor_dim1 lo=512
    g1[3] = __builtin_amdgcn_readfirstlane((int)(32u  << 16)); // dim1 hi=0 | tile_dim0=32
    g1[4] = __builtin_amdgcn_readfirstlane(16);                // tile_dim1=16, tile_dim2=0
    g1[5] = __builtin_amdgcn_readfirstlane(512);               // tensor_dim0_stride[31:0]
    g1[6] = 0;                                                 // stride hi, dim1_stride lo
    g1[7] = 0;
    const i32x4 z = {0, 0, 0, 0};
#if defined(__clang_major__) && __clang_major__ >= 23
    const i32x8 z8 = {0, 0, 0, 0, 0, 0, 0, 0};
    __builtin_amdgcn_tensor_load_to_lds(g0, g1, z, z, z8, 0);
#else
    __builtin_amdgcn_tensor_load_to_lds(g0, g1, z, z, 0);
#endif
}
#endif

// ---------------------------------------------------------------- main pairwise kernel
// One wave computes one 16x16 (i,j) tile, only for tileJ >= tileI (symmetry):
//   P_ij = 0.5*(sq_i + sq_j) - Gram_ij - 0.5 * sum_k relu(|d|-1)^2
// Gram via V_WMMA_F32_16X16X4_F32, correction via packed VALU, fed from LDS.
__global__ __launch_bounds__(128) void pair_kernel(const float* __restrict__ T,
                                                   const float* __restrict__ S,
                                                   float* __restrict__ ws) {
    __shared__ float lds[4 * 2 * 16 * LROW];   // 4 waves * (A,B) * 16 rows * LROW

    const int mat = blockIdx.y;
    const float* __restrict__ X  = mat ? S : T;
    const float* __restrict__ sq = ws + WS_SQ + mat * NN;
    float* __restrict__ P        = ws + WS_P  + (size_t)mat * NN * NN;
    float* __restrict__ tot      = ws + WS_TOT + mat;

    const int lane = threadIdx.x & 31;
    const int wave = threadIdx.x >> 5;
    const int tile = blockIdx.x * 4 + wave;    // 0..527 (upper-triangular tiles)

    // map linear upper-triangular index -> (tI, tJ), tJ >= tI
    int tI = 0, rem = tile;
    while (rem >= (32 - tI)) { rem -= (32 - tI); ++tI; }
    const int tJ = tI + rem;
    const int i0 = tI * 16;
    const int j0 = tJ * 16;

    float* ldsA = lds + wave * (2 * 16 * LROW);
    float* ldsB = ldsA + 16 * LROW;

    const int h = lane >> 4;                   // wave half (0/1)
    const int n = lane & 15;                   // output column within tile

    v8f acc = {};                              // WMMA C/D accumulator (Gram tile)
    float corr[8];
#pragma unroll
    for (int r = 0; r < 8; ++r) corr[r] = 0.f;

#if defined(HAVE_TDM)
    const unsigned ldsAb = (unsigned)(size_t)(void*)ldsA;
    const unsigned ldsBb = (unsigned)(size_t)(void*)ldsB;
#endif

    for (int k0 = 0; k0 < DD; k0 += KC) {
#if defined(HAVE_TDM)
        // previous chunk's LDS reads must have landed before DMA overwrites slice
        asm volatile("s_wait_dscnt 0x0" ::: "memory");
        tdm_load_tile(X + (size_t)i0 * DD + k0, ldsAb);
        tdm_load_tile(X + (size_t)j0 * DD + k0, ldsBb);
        asm volatile("" ::: "memory");
        __builtin_amdgcn_s_wait_tensorcnt(0);
        asm volatile("" ::: "memory");
#else
        __syncthreads();                       // protect previous chunk reads (WAR)
#pragma unroll
        for (int q = 0; q < 4; ++q) {
            const int f4  = lane + 32 * q;     // float4 slot 0..127
            const int row = f4 >> 3;           // 16 rows
            const int kk4 = (f4 & 7) * 4;      // 8 float4 per row
            const float4 ga = *(const float4*)(X + (size_t)(i0 + row) * DD + k0 + kk4);
            const float4 gb = *(const float4*)(X + (size_t)(j0 + row) * DD + k0 + kk4);
            *(float4*)&ldsA[row * LROW + kk4] = ga;
            *(float4*)&ldsB[row * LROW + kk4] = gb;
            if (k0 + KC < DD) {               // CDNA5 global_prefetch_b8 of next chunk
                __builtin_prefetch(X + (size_t)(i0 + row) * DD + k0 + KC + kk4, 0, 3);
                __builtin_prefetch(X + (size_t)(j0 + row) * DD + k0 + KC + kk4, 0, 3);
            }
        }
        __syncthreads();
#endif

        // ---- Gram accumulation: 8 x V_WMMA_F32_16X16X4_F32 over this chunk.
        // A 16x4 layout: lanes 0-15 rows M, vgpr{0,1}=K{0,1}; lanes 16-31 K{2,3}.
        // B 4x16 layout mirrors it with B[k][n] = X[j0+n][k].
#pragma unroll
        for (int s = 0; s < KC / 4; ++s) {
            const int koff = s * 4 + 2 * h;
            v2f a = *(const v2f*)&ldsA[n * LROW + koff];
            v2f b = *(const v2f*)&ldsB[n * LROW + koff];
            acc = __builtin_amdgcn_wmma_f32_16x16x4_f32(
                false, a, false, b, (short)0, acc, false, false);
        }

        // ---- correction: sum_k relu(|a_ik - b_jk| - 1)^2 in the WMMA C mapping
        //      (vgpr r, lane) -> row i0 + 8*h + r, col j0 + n
#pragma unroll
        for (int c = 0; c < KC / 4; ++c) {
            const int kk = c * 4;
            const float4 bv = *(const float4*)&ldsB[n * LROW + kk];
#pragma unroll
            for (int r = 0; r < 8; ++r) {
                const float4 av = *(const float4*)&ldsA[(8 * h + r) * LROW + kk];
                float t;
                t = fmaxf(fabsf(av.x - bv.x) - 1.f, 0.f); corr[r] = fmaf(t, t, corr[r]);
                t = fmaxf(fabsf(av.y - bv.y) - 1.f, 0.f); corr[r] = fmaf(t, t, corr[r]);
                t = fmaxf(fabsf(av.z - bv.z) - 1.f, 0.f); corr[r] = fmaf(t, t, corr[r]);
                t = fmaxf(fabsf(av.w - bv.w) - 1.f, 0.f); corr[r] = fmaf(t, t, corr[r]);
            }
        }
    }

    // ---- combine, store pairwise values (+mirror), accumulate total for the mean
    const float sqj  = sq[j0 + n];
    const bool  offd = (tI != tJ);
    float lsum = 0.f;
#pragma unroll
    for (int r = 0; r < 8; ++r) {
        const int i = i0 + 8 * h + r;
        const float val = 0.5f * (sq[i] + sqj) - acc[r] - 0.5f * corr[r];
        P[(size_t)i * NN + (j0 + n)] = val;
        if (offd) P[(size_t)(j0 + n) * NN + i] = val;   // symmetric mirror
        lsum += offd ? 2.f * val : val;
    }
    for (int off = 16; off > 0; off >>= 1) lsum += __shfl_xor(lsum, off, 32);
    if (lane == 0) atomicAdd(tot, lsum);
}

// ---------------------------------------------------------------- final reduction
__global__ __launch_bounds__(256) void final_kernel(const float* __restrict__ ws,
                                                    float* __restrict__ out) {
    const float St = ws[WS_TOT + 0];
    const float Ss = ws[WS_TOT + 1];
    const float n2 = (float)NN * (float)NN;
    const float ct = n2 / St;                  // 1/mean_t
    const float cs = n2 / Ss;                  // 1/mean_s
    const float* __restrict__ Pt = ws + WS_P;
    const float* __restrict__ Ps = ws + WS_P + (size_t)NN * NN;

    float s = 0.f;
    for (int idx = blockIdx.x * blockDim.x + threadIdx.x; idx < NN * NN;
         idx += gridDim.x * blockDim.x)
        s += fabsf(Pt[idx] * ct - Ps[idx] * cs);

    __shared__ float red[256];
    red[threadIdx.x] = s;
    __syncthreads();
    for (int off = 128; off > 0; off >>= 1) {
        if (threadIdx.x < off) red[threadIdx.x] += red[threadIdx.x + off];
        __syncthreads();
    }
    if (threadIdx.x == 0) atomicAdd(out, red[0]);
}

// ---------------------------------------------------------------- launch
extern "C" void kernel_launch(void* const* d_in, const int* in_sizes, int n_in,
                              void* d_out, int out_size, void* d_ws, size_t ws_size,
                              hipStream_t stream) {
    const float* T = (const float*)d_in[0];
    const float* S = (const float*)d_in[1];
    float* ws  = (float*)d_ws;
    float* out = (float*)d_out;

    hipLaunchKernelGGL(init_kernel,    dim3(1),      dim3(1),   0, stream, ws, out);
    hipLaunchKernelGGL(rownorm_kernel, dim3(1024),   dim3(128), 0, stream, T, S, ws);
    // 528 upper-triangular tiles / 4 waves per block = 132 blocks, y = matrix
    hipLaunchKernelGGL(pair_kernel,    dim3(132, 2), dim3(128), 0, stream, T, S, ws);
    hipLaunchKernelGGL(final_kernel,   dim3(256),    dim3(256), 0, stream, ws, out);
}